// SeriesSoap_56470230007940
// MI455X (gfx1250) — compile-verified
//
#include <hip/hip_runtime.h>
#include <cmath>

#define LSZ   7      // LMAX+1
#define NM    6      // NMAX+1 radial channels
#define NCELL 49     // L*L packed (l,m) cells
#define RCUT  6.0f
#define K00   0.28209479177387814f

typedef __attribute__((ext_vector_type(2))) float v2f;
typedef __attribute__((ext_vector_type(8))) float v8f;

struct Q { float v, x, y, z; };   // value + d/dx,d/dy,d/dz (forward-mode AD)

// ---------------------------------------------------------------------------
// radial: f[a] = (1-d/rc)^2 * d^(2a),  df[a] = d f[a]/dd
// ---------------------------------------------------------------------------
__device__ __forceinline__ void radial_fn(float d, float d2, float* fv, float* dfr) {
    float step = (d < RCUT) ? 1.f : 0.f;
    float t    = 1.f - d * (1.f / RCUT);
    float r    = t * t * step;
    float dr   = (-2.f / RCUT) * t * step;
    float invd = 1.f / d;
    float dp   = 1.f;
#pragma unroll
    for (int a = 0; a < NM; ++a) {
        fv[a]  = r * dp;
        dfr[a] = dr * dp + r * (2.f * (float)a) * dp * invd;
        dp *= d2;
    }
}

// ---------------------------------------------------------------------------
// Zero the 294-float c accumulator in workspace (run each call — harness
// poisons ws once and never re-poisons; we must own initialization).
// ---------------------------------------------------------------------------
__global__ void soap_zero_c(float* __restrict__ cws) {
    int t = threadIdx.x;
    if (t < NM * NCELL) cws[t] = 0.f;
}

// ---------------------------------------------------------------------------
// Pass 1: c[n][cell] = sum_j f[n,j] * Ypack[cell,j]   via V_WMMA_F32_16X16X4_F32
//   GEMM:  C(16x64 padded) = A(16 x K) * B(K x 64),  K = atoms, 4 per step.
//   A rows 0..5 = radial channels, B cols 0..48 = packed lm cells.
// ---------------------------------------------------------------------------
__global__ void __launch_bounds__(256) soap_pass1(const float* __restrict__ xyz,
                                                  float* __restrict__ cws, int N) {
    __shared__ float smem[256 * NM + 256 * NCELL];   // 14080 floats = 56.3 KB
    float* Fl  = smem;               // [atom][6]
    float* Yl  = smem + 256 * NM;    // [atom][49]
    float* Red = smem;               // overlay, used only after tile loop

    const int tid  = threadIdx.x;
    const int lane = tid & 31;
    const int w    = tid >> 5;
    const int hi   = lane >> 4;
    const int M    = lane & 15;

    v8f acc[4] = {};                 // 4 N-tiles of the 16x16 f32 accumulator

    const int numTiles = (N + 255) >> 8;
    for (int tile = blockIdx.x; tile < numTiles; tile += gridDim.x) {
        const int j = (tile << 8) + tid;
        float px = 0.f, py = 0.f, pz = 0.f;
        if (j < N) { px = xyz[3 * j]; py = xyz[3 * j + 1]; pz = xyz[3 * j + 2]; }
        const float d2 = px * px + py * py + pz * pz;
        const float d  = sqrtf(d2);

        float fv[NM], dfr[NM];
        radial_fn(d, d2, fv, dfr);
        if (j >= N) {
#pragma unroll
            for (int a = 0; a < NM; ++a) fv[a] = 0.f;   // kill padded lanes
        }
#pragma unroll
        for (int a = 0; a < NM; ++a) Fl[tid * NM + a] = fv[a];

        // ---- packed solid harmonics (values only), streamed into LDS ----
        float reP[7] = {}, imP[7] = {}, reP2[7] = {}, imP2[7] = {};
        float reC[7] = {}, imC[7] = {};
        float* myY = Yl + tid * NCELL;
        reC[0] = K00;
        myY[0] = reC[0];
#pragma unroll
        for (int l = 1; l < LSZ; ++l) {
#pragma unroll
            for (int m = 0; m < 7; ++m) { reP2[m] = reP[m]; imP2[m] = imP[m];
                                          reP[m]  = reC[m]; imP[m]  = imC[m]; }
            const float al = -sqrtf((2.f * l + 1.f) / (2.f * l));
            reC[l] = al * (px * reP[l - 1] - py * imP[l - 1]);
            imC[l] = al * (px * imP[l - 1] + py * reP[l - 1]);
#pragma unroll
            for (int m = 0; m < l; ++m) {
                const float Ac   = sqrtf((float)(l * l - m * m) /
                                         (float)((2 * l - 1) * (2 * l + 1)));
                const float invA = 1.f / Ac;
                if (l - 2 >= m) {
                    const float Ap = sqrtf((float)((l - 1) * (l - 1) - m * m) /
                                           (float)((2 * l - 3) * (2 * l - 1)));
                    reC[m] = (pz * reP[m] - Ap * d2 * reP2[m]) * invA;
                    imC[m] = (pz * imP[m] - Ap * d2 * imP2[m]) * invA;
                } else {
                    reC[m] = pz * reP[m] * invA;
                    imC[m] = pz * imP[m] * invA;
                }
            }
#pragma unroll
            for (int jj = 0; jj <= l; ++jj) myY[l * 7 + jj] = reC[l - jj];
#pragma unroll
            for (int ii = 0; ii <  l; ++ii) myY[ii * 7 + l] = imC[ii + 1];
        }
        __syncthreads();

        // ---- WMMA accumulate: this wave consumes its 32 atoms, K=4/step ----
        const int base = w << 5;
#pragma unroll
        for (int kk = 0; kk < 8; ++kk) {
            const int a0 = base + kk * 4 + (hi ? 2 : 0);
            const int a1 = base + kk * 4 + (hi ? 3 : 1);
            v2f Av;
            Av.x = (M < NM) ? Fl[a0 * NM + M] : 0.f;
            Av.y = (M < NM) ? Fl[a1 * NM + M] : 0.f;
#pragma unroll
            for (int t = 0; t < 4; ++t) {
                const int col = t * 16 + M;
                v2f Bv;
                Bv.x = (col < NCELL) ? Yl[a0 * NCELL + col] : 0.f;
                Bv.y = (col < NCELL) ? Yl[a1 * NCELL + col] : 0.f;
                acc[t] = __builtin_amdgcn_wmma_f32_16x16x4_f32(
                    false, Av, false, Bv, (short)0, acc[t], false, false);
            }
        }
        __syncthreads();
    }

    // ---- block reduction of the 8 waves' partial C tiles, then atomicAdd ----
    if (hi == 0) {   // lanes 0..15 hold rows 0..7; only rows(VGPR) 0..5 are live
#pragma unroll
        for (int t = 0; t < 4; ++t) {
#pragma unroll
            for (int r = 0; r < 6; ++r)
                Red[w * 384 + t * 96 + r * 16 + M] = acc[t][r];
        }
    }
    __syncthreads();
    if (tid < 384) {
        float sum = 0.f;
#pragma unroll
        for (int w2 = 0; w2 < 8; ++w2) sum += Red[w2 * 384 + tid];
        const int t    = tid / 96;
        const int r    = (tid % 96) / 16;
        const int colL = tid % 16;
        const int col  = t * 16 + colL;
        if (col < NCELL) atomicAdd(&cws[r * NCELL + col], sum);
    }
}

// ---------------------------------------------------------------------------
// p[q,a,b] = sum over cells with max(i,j)==q of w * c[a][cell] * c[b][cell]
// ---------------------------------------------------------------------------
__global__ void soap_p(const float* __restrict__ cws, float* __restrict__ out) {
    __shared__ float cs[NM * NCELL];
    const int tid = threadIdx.x;
    if (tid < NM * NCELL) cs[tid] = cws[tid];
    __syncthreads();
    if (tid >= LSZ * NM * NM) return;
    const int q = tid / 36;
    const int a = (tid % 36) / 6;
    const int b = tid % 6;
    float sum = 0.f;
    for (int jj = 0; jj <= q; ++jj) {
        const float wgt = (jj == q) ? 1.f : 2.f;
        const int cell  = q * 7 + jj;
        sum += wgt * cs[a * NCELL + cell] * cs[b * NCELL + cell];
    }
    for (int ii = 0; ii < q; ++ii) {
        const int cell = ii * 7 + q;
        sum += 2.f * cs[a * NCELL + cell] * cs[b * NCELL + cell];
    }
    out[tid] = sum;
}

// ---------------------------------------------------------------------------
// Pass 3 (dominant, store-bound ~198MB): per-atom forward-mode AD of the
// harmonic recurrence; per l-row compute G_l[b], H_l[b][xyz] and stream the
// (l,a,b) dp plane with non-temporal b96-wide coalesced stores.
// ---------------------------------------------------------------------------
__global__ void __launch_bounds__(256) soap_dp(const float* __restrict__ xyz,
                                               const float* __restrict__ cws,
                                               float* __restrict__ out, int N) {
    __shared__ float cs[NM * NCELL];
    const int tid = threadIdx.x;
    if (tid < NM * NCELL) cs[tid] = cws[tid];
    __syncthreads();

    const int j = blockIdx.x * 256 + tid;
    if (j >= N) return;

    const float px = xyz[3 * j], py = xyz[3 * j + 1], pz = xyz[3 * j + 2];
    const float d2   = px * px + py * py + pz * pz;
    const float d    = sqrtf(d2);
    const float invd = 1.f / d;
    const float s0 = px * invd, s1 = py * invd, s2 = pz * invd;

    float fv[NM], dfr[NM];
    radial_fn(d, d2, fv, dfr);

    float* dpo = out + LSZ * NM * NM;   // dp section starts after p (252 floats)

    Q reP[7] = {}, imP[7] = {}, reP2[7] = {}, imP2[7] = {};
    Q reC[7] = {}, imC[7] = {};
    reC[0].v = K00;

#pragma unroll
    for (int l = 0; l < LSZ; ++l) {
        if (l > 0) {
#pragma unroll
            for (int m = 0; m < 7; ++m) { reP2[m] = reP[m]; imP2[m] = imP[m];
                                          reP[m]  = reC[m]; imP[m]  = imC[m]; }
            {   // sectoral m = l
                const float al = -sqrtf((2.f * l + 1.f) / (2.f * l));
                const Q rp = reP[l - 1], ip = imP[l - 1];
                reC[l].v = al * (px * rp.v - py * ip.v);
                reC[l].x = al * (rp.v + px * rp.x - py * ip.x);
                reC[l].y = al * (px * rp.y - ip.v - py * ip.y);
                reC[l].z = al * (px * rp.z - py * ip.z);
                imC[l].v = al * (px * ip.v + py * rp.v);
                imC[l].x = al * (ip.v + px * ip.x + py * rp.x);
                imC[l].y = al * (px * ip.y + rp.v + py * rp.y);
                imC[l].z = al * (px * ip.z + py * rp.z);
            }
#pragma unroll
            for (int m = 0; m < l; ++m) {
                const float Ac   = sqrtf((float)(l * l - m * m) /
                                         (float)((2 * l - 1) * (2 * l + 1)));
                const float invA = 1.f / Ac;
                const Q r1 = reP[m], i1 = imP[m];
                if (l - 2 >= m) {
                    const float Ap = sqrtf((float)((l - 1) * (l - 1) - m * m) /
                                           (float)((2 * l - 3) * (2 * l - 1)));
                    const Q r2 = reP2[m], i2 = imP2[m];
                    reC[m].v = (pz * r1.v - Ap * d2 * r2.v) * invA;
                    reC[m].x = (pz * r1.x - Ap * (2.f * px * r2.v + d2 * r2.x)) * invA;
                    reC[m].y = (pz * r1.y - Ap * (2.f * py * r2.v + d2 * r2.y)) * invA;
                    reC[m].z = (r1.v + pz * r1.z - Ap * (2.f * pz * r2.v + d2 * r2.z)) * invA;
                    imC[m].v = (pz * i1.v - Ap * d2 * i2.v) * invA;
                    imC[m].x = (pz * i1.x - Ap * (2.f * px * i2.v + d2 * i2.x)) * invA;
                    imC[m].y = (pz * i1.y - Ap * (2.f * py * i2.v + d2 * i2.y)) * invA;
                    imC[m].z = (i1.v + pz * i1.z - Ap * (2.f * pz * i2.v + d2 * i2.z)) * invA;
                } else {
                    reC[m].v = pz * r1.v * invA;
                    reC[m].x = pz * r1.x * invA;
                    reC[m].y = pz * r1.y * invA;
                    reC[m].z = (r1.v + pz * r1.z) * invA;
                    imC[m].v = pz * i1.v * invA;
                    imC[m].x = pz * i1.x * invA;
                    imC[m].y = pz * i1.y * invA;
                    imC[m].z = (i1.v + pz * i1.z) * invA;
                }
            }
        }

        // ---- consume row l: all cells with max(i,j)==l are now available ----
        float Gl[NM] = {}, Hx[NM] = {}, Hy[NM] = {}, Hz[NM] = {};
#pragma unroll
        for (int jj = 0; jj <= l; ++jj) {               // cell (l, jj) = Re(l, l-jj)
            const Q q = reC[l - jj];
            const float wgt = (jj == l) ? 1.f : 2.f;
            const int cell  = l * 7 + jj;
#pragma unroll
            for (int b = 0; b < NM; ++b) {
                const float wc = wgt * cs[b * NCELL + cell];
                Gl[b] += wc * q.v; Hx[b] += wc * q.x;
                Hy[b] += wc * q.y; Hz[b] += wc * q.z;
            }
        }
#pragma unroll
        for (int ii = 0; ii < l; ++ii) {                // cell (ii, l) = Im(l, ii+1)
            const Q q = imC[ii + 1];
            const int cell = ii * 7 + l;
#pragma unroll
            for (int b = 0; b < NM; ++b) {
                const float wc = 2.f * cs[b * NCELL + cell];
                Gl[b] += wc * q.v; Hx[b] += wc * q.x;
                Hy[b] += wc * q.y; Hz[b] += wc * q.z;
            }
        }

        // ---- stream dp plane q=l :  dp = T + T^T  (non-temporal, coalesced) ----
#pragma unroll
        for (int a = 0; a < NM; ++a) {
#pragma unroll
            for (int b = 0; b < NM; ++b) {
                const float u  = dfr[a] * Gl[b] + dfr[b] * Gl[a];
                const float o0 = s0 * u + fv[a] * Hx[b] + fv[b] * Hx[a];
                const float o1 = s1 * u + fv[a] * Hy[b] + fv[b] * Hy[a];
                const float o2 = s2 * u + fv[a] * Hz[b] + fv[b] * Hz[a];
                const size_t off =
                    (((size_t)(l * 36 + a * 6 + b) * (size_t)N) + (size_t)j) * 3;
                __builtin_nontemporal_store(o0, dpo + off);
                __builtin_nontemporal_store(o1, dpo + off + 1);
                __builtin_nontemporal_store(o2, dpo + off + 2);
            }
        }
    }
}

// ---------------------------------------------------------------------------
extern "C" void kernel_launch(void* const* d_in, const int* in_sizes, int n_in,
                              void* d_out, int out_size, void* d_ws, size_t ws_size,
                              hipStream_t stream) {
    (void)n_in; (void)out_size; (void)ws_size;
    const float* xyz = (const float*)d_in[0];
    float* out = (float*)d_out;
    float* cws = (float*)d_ws;                 // 294-float c accumulator
    const int N = in_sizes[0] / 3;

    soap_zero_c<<<1, 256, 0, stream>>>(cws);
    soap_pass1 <<<128, 256, 0, stream>>>(xyz, cws, N);
    soap_p     <<<1, 256, 0, stream>>>(cws, out);
    soap_dp    <<<(N + 255) / 256, 256, 0, stream>>>(xyz, cws, out, N);
}